// KaolinRenderer_15401752723827
// MI455X (gfx1250) — compile-verified
//
#include <hip/hip_runtime.h>
#include <stdint.h>

typedef __attribute__((ext_vector_type(2))) float v2f;
typedef __attribute__((ext_vector_type(4))) float v4f;
typedef __attribute__((ext_vector_type(8))) float v8f;

#define IMG   256
#define NB    2
#define NV    4096
#define NF    4096
#define TILES 16                       // 16x16 tiles of 16x16 pixels
#define SLICES 8
#define FACES_PER_SLICE (NF / SLICES)  // 512
#define GROUPS_PER_SLICE (FACES_PER_SLICE / 4)  // 128 groups of 4 faces
#define GBATCH 8                       // groups culled per ballot (32 faces)

// FOV 60deg -> f = 1/tan(30deg)
#define FOVF 1.7320508075688772f

// ---------------- Kernel 1: vertex transform to NDC (x,y only; z is dead) ---
__global__ void k_xform(const float* __restrict__ verts,
                        const float* __restrict__ Rm,
                        const float* __restrict__ Tm,
                        float* __restrict__ ndc) {
  int tid = blockIdx.x * blockDim.x + threadIdx.x;
  if (tid >= NB * NV) return;
  int b = tid / NV;
  const float* Rb = Rm + b * 9;
  const float* Tb = Tm + b * 3;
  float x = verts[tid * 3 + 0];
  float y = verts[tid * 3 + 1];
  float z = verts[tid * 3 + 2];
  float dx = x - Tb[0], dy = y - Tb[1], dz = z - Tb[2];
  float cx = Rb[0] * dx + Rb[3] * dy + Rb[6] * dz;
  float cy = Rb[1] * dx + Rb[4] * dy + Rb[7] * dz;
  float cw = -(Rb[2] * dx + Rb[5] * dy + Rb[8] * dz);
  float iw = 1.0f / fmaxf(cw, 1e-8f);
  ndc[tid * 2 + 0] = FOVF * cx * iw;
  ndc[tid * 2 + 1] = FOVF * cy * iw;
}

// ---------------- Kernel 2: per-face edge coefficients + bbox; zero masks ---
// fdata layout (16 floats/face): [A0,B0,C0, A1,B1,C1, A2,B2,C2, 0, minx,maxx,miny,maxy, 0,0]
__global__ void k_face(const float* __restrict__ ndc,
                       const int* __restrict__ faces,
                       float* __restrict__ fdata,
                       unsigned* __restrict__ masks) {
  int tid = blockIdx.x * blockDim.x + threadIdx.x;
  if (tid < NB * TILES * TILES * 16) masks[tid] = 0u;   // re-zero every call
  if (tid >= NB * NF) return;
  int b = tid / NF;
  int i0 = faces[tid * 3 + 0];
  int i1 = faces[tid * 3 + 1];
  int i2 = faces[tid * 3 + 2];
  const float* nb = ndc + (size_t)b * NV * 2;
  float x0 = nb[i0 * 2 + 0], y0 = nb[i0 * 2 + 1];
  float x1 = nb[i1 * 2 + 0], y1 = nb[i1 * 2 + 1];
  float x2 = nb[i2 * 2 + 0], y2 = nb[i2 * 2 + 1];
  // e = A*px + B*py + C  (matches reference edge functions)
  float A0 = -(y1 - y0), B0 = (x1 - x0), C0 = (y1 - y0) * x0 - (x1 - x0) * y0;
  float A1 = -(y2 - y1), B1 = (x2 - x1), C1 = (y2 - y1) * x1 - (x2 - x1) * y1;
  float A2 = -(y0 - y2), B2 = (x0 - x2), C2 = (y0 - y2) * x2 - (x0 - x2) * y2;
  float mnx = fminf(x0, fminf(x1, x2)), mxx = fmaxf(x0, fmaxf(x1, x2));
  float mny = fminf(y0, fminf(y1, y2)), mxy = fmaxf(y0, fmaxf(y1, y2));
  v4f* o = (v4f*)(fdata + (size_t)tid * 16);
  v4f q0; q0.x = A0;  q0.y = B0;  q0.z = C0;  q0.w = A1;
  v4f q1; q1.x = B1;  q1.y = C1;  q1.z = A2;  q1.w = B2;
  v4f q2; q2.x = C2;  q2.y = 0.f; q2.z = mnx; q2.w = mxx;
  v4f q3; q3.x = mny; q3.y = mxy; q3.z = 0.f; q3.w = 0.f;
  o[0] = q0; o[1] = q1; o[2] = q2; o[3] = q3;
}

// per-lane score for its two triangles in a 16x16x4 WMMA result:
// inside(tri) <=> min3(e) >= 0 || max3(e) <= 0 <=> max(min3(e), -max3(e)) >= 0
// OR of the 2 triangles <=> max of their scores
__device__ __forceinline__ float tri_score(v8f c) {
  float m0  = fminf(fminf(c[0], c[1]), c[2]);
  float M0n = fminf(fminf(-c[0], -c[1]), -c[2]);   // = -max3
  float m1  = fminf(fminf(c[3], c[4]), c[5]);
  float M1n = fminf(fminf(-c[3], -c[4]), -c[5]);
  float t   = fmaxf(m1, M1n);
  return fmaxf(fmaxf(m0, M0n), t);                 // -> v_max3
}

// ---------------- Kernel 3: WMMA rasterizer -------------------------------
// grid: NB*TILES*TILES blocks x 256 threads (8 waves); wave = one face slice.
// WMMA f32 16x16x4:  D[edge_row, pixel] = A_e*px + B_e*py + C_e*1 + 0
//   A rows 0-2 tri0, 3-5 tri1, 6-7 pad, 8-10 tri2, 11-13 tri3, 14-15 pad.
__global__ void __launch_bounds__(256) k_raster(const float* __restrict__ fdata,
                                                unsigned* __restrict__ masks) {
  int block = blockIdx.x;
  int tx = block & (TILES - 1);
  int ty = (block >> 4) & (TILES - 1);
  int b  = block >> 8;
  int slice = threadIdx.x >> 5;
  int lane  = threadIdx.x & 31;
  int col   = lane & 15;
  bool hi   = lane >= 16;

  const float scale = 2.0f / IMG;
  float pxv   = ((float)(tx * 16 + col) + 0.5f) * scale - 1.0f;
  float tminx = (float)(tx * 16) * scale - 1.0f;
  float tmaxx = (float)(tx * 16 + 16) * scale - 1.0f;
  float tminy = (float)(ty * 16) * scale - 1.0f;
  float tmaxy = (float)(ty * 16 + 16) * scale - 1.0f;

  // B-matrix fragments: loop-invariant -> keep resident in VGPR pairs.
  // lanes 0-15: (px, py_yb) = (K0,K1); lanes 16-31: (1, 0) = (K2,K3)
  v2f bmat[16];
#pragma unroll
  for (int yb = 0; yb < 16; ++yb) {
    float pyv = ((float)(ty * 16 + yb) + 0.5f) * scale - 1.0f;
    bmat[yb].x = hi ? 1.0f : pxv;
    bmat[yb].y = hi ? 0.0f : pyv;
  }

  // per-row running coverage score: covered(yb) <=> q[yb] >= 0 at the end
  float q[16];
#pragma unroll
  for (int yb = 0; yb < 16; ++yb) q[yb] = -__builtin_inff();

  // A-matrix row -> (face-in-group, edge) mapping
  int r = col;
  bool valid = (r < 6) || (r >= 8 && r < 14);
  int lf = (r < 6) ? (r / 3) : (2 + (r - 8) / 3);
  int e  = (r < 6) ? (r % 3) : ((r - 8) % 3);

  const float* fb = fdata + (size_t)b * NF * 16;

  for (int batch = 0; batch < GROUPS_PER_SLICE / GBATCH; ++batch) {  // 16 iters
    int fbase = slice * FACES_PER_SLICE + batch * (GBATCH * 4);
    // 32 lanes test 32 face bboxes (8 groups) -> one ballot, uniform mask
    const float* bbp = fb + (size_t)(fbase + lane) * 16 + 10;
    bool ov = (bbp[0] <= tmaxx) & (bbp[1] >= tminx) &
              (bbp[2] <= tmaxy) & (bbp[3] >= tminy);
    unsigned gm = __builtin_amdgcn_ballot_w32(ov);
    if (gm == 0u) continue;

    if (batch + 1 < GROUPS_PER_SLICE / GBATCH)
      __builtin_prefetch(fb + (size_t)(fbase + GBATCH * 4) * 16, 0, 3);  // WGP scope

#pragma unroll 1
    for (int gg = 0; gg < GBATCH; ++gg) {
      if (((gm >> (gg * 4)) & 0xFu) == 0u) continue;   // uniform s_cbranch
      int face0 = fbase + gg * 4;

      // A fragment: lanes 0-15 -> (A,B)=(K0,K1); lanes 16-31 -> (C,0)=(K2,K3)
      v2f a;
      if (!valid) {
        a.x = 0.f; a.y = 0.f;
      } else {
        const float* ep = fb + (size_t)(face0 + lf) * 16 + e * 3;
        if (!hi) { a.x = ep[0]; a.y = ep[1]; }
        else     { a.x = ep[2]; a.y = 0.f;  }
      }

      // paired independent accumulators -> WMMAs issue back-to-back and
      // the VALU score of pair-member 0 co-executes with WMMA of member 1
#pragma unroll
      for (int yb = 0; yb < 16; yb += 2) {
        v8f c0 = {0.f, 0.f, 0.f, 0.f, 0.f, 0.f, 0.f, 0.f};
        v8f c1 = {0.f, 0.f, 0.f, 0.f, 0.f, 0.f, 0.f, 0.f};
        c0 = __builtin_amdgcn_wmma_f32_16x16x4_f32(
            false, a, false, bmat[yb], (short)0, c0, false, false);
        c1 = __builtin_amdgcn_wmma_f32_16x16x4_f32(
            false, a, false, bmat[yb + 1], (short)0, c1, false, false);
        q[yb]     = fmaxf(q[yb],     tri_score(c0));
        q[yb + 1] = fmaxf(q[yb + 1], tri_score(c1));
      }
    }
  }

  // epilogue: scores -> 16-bit row mask
  unsigned cov = 0u;
#pragma unroll
  for (int yb = 0; yb < 16; ++yb)
    if (q[yb] >= 0.0f) cov |= (1u << yb);

  // lanes L and L^16 hold coverage for the same pixel column: SWAPX16 + OR
  cov |= (unsigned)__builtin_amdgcn_ds_swizzle((int)cov, 0x401F);
  if (!hi) {
    unsigned* mp = masks + ((((size_t)b * TILES + ty) * TILES + tx) * 16 + col);
    atomicOr(mp, cov);
  }
}

// ---------------- Kernel 4: expand bitmask -> [B,H,W,3] float --------------
__global__ void k_expand(const unsigned* __restrict__ masks,
                         float* __restrict__ out) {
  int tid = blockIdx.x * blockDim.x + threadIdx.x;
  if (tid >= NB * IMG * IMG) return;
  int b = tid >> 16;
  int p = tid & 0xFFFF;
  int y = p >> 8;
  int x = p & 255;
  unsigned m = masks[(((size_t)b * TILES + (y >> 4)) * TILES + (x >> 4)) * 16 + (x & 15)];
  float v = ((m >> (y & 15)) & 1u) ? 1.0f : 0.0f;
  float* o = out + (size_t)tid * 3;
  o[0] = v; o[1] = v; o[2] = v;
}

// ---------------- Launch ----------------------------------------------------
extern "C" void kernel_launch(void* const* d_in, const int* in_sizes, int n_in,
                              void* d_out, int out_size, void* d_ws, size_t ws_size,
                              hipStream_t stream) {
  (void)in_sizes; (void)n_in; (void)out_size; (void)ws_size;
  const float* verts = (const float*)d_in[0];
  const int*   faces = (const int*)d_in[1];
  const float* Rm    = (const float*)d_in[2];
  const float* Tm    = (const float*)d_in[3];

  float* ws    = (float*)d_ws;
  float* ndc   = ws;                          // NB*NV*2 floats   (64 KB)
  float* fdata = ws + (size_t)NB * NV * 2;    // NB*NF*16 floats  (512 KB)
  unsigned* masks = (unsigned*)(fdata + (size_t)NB * NF * 16);  // NB*4096 u32 (32 KB)
  float* out = (float*)d_out;

  k_xform <<<(NB * NV + 255) / 256, 256, 0, stream>>>(verts, Rm, Tm, ndc);
  k_face  <<<(NB * NF + 255) / 256, 256, 0, stream>>>(ndc, faces, fdata, masks);
  k_raster<<<NB * TILES * TILES, 256, 0, stream>>>(fdata, masks);
  k_expand<<<(NB * IMG * IMG + 255) / 256, 256, 0, stream>>>(masks, out);
}